// PairwisePotential1_25812753448975
// MI455X (gfx1250) — compile-verified
//
#include <hip/hip_runtime.h>

// Problem constants (match the reference: K=3, P=1, C=64, S=128, B=16)
#define S_DIM     128
#define SP        130                 // padded plane width  (S + 2)
#define PLANE     (S_DIM * S_DIM)     // 16384
#define TROWS     64                  // output rows handled per block
#define TR_TILE   (TROWS + 2)         // 66 padded-space rows resident in LDS
#define TILE_EL   (TR_TILE * SP)      // 8580 floats = 34,320 B of LDS
#define NTHREADS  256                 // 8 wave32 per block

// Low 32 bits of a generic pointer to __shared__ == workgroup-relative LDS
// byte offset (flat->LDS aperture truncates to addr[31:0]; the async-to-LDS
// instruction adds LDS_BASE itself).
__device__ __forceinline__ unsigned lds_addr32(const void* p) {
  return (unsigned)(unsigned long long)p;
}

__global__ __launch_bounds__(NTHREADS)
void pairwise_potential_kernel(const float* __restrict__ x,
                               const float* __restrict__ w1,
                               const float* __restrict__ w2,
                               float* __restrict__ out,
                               int nch)
{
  __shared__ float tile[TILE_EL];

  const int tid = threadIdx.x;
  const int blk = blockIdx.x;
  const int p   = blk >> 1;          // plane index = b*C + c
  const int h   = blk & 1;           // which row-half of the plane
  const int c   = p % nch;           // channel for the broadcast weights
  const int y0  = h * TROWS;         // base row (padded-space == output-space)

  const float* xplane = x  + (size_t)p * PLANE;
  const float* w1o    = w1 + (size_t)c * PLANE + (size_t)y0 * S_DIM;
  const float* w2o    = w2 + (size_t)c * PLANE + (size_t)y0 * S_DIM;
  float*       op     = out + (size_t)p * PLANE + (size_t)y0 * S_DIM;

  // ---- Phase 1: zero the LDS tile (provides the zero-padding halo) --------
  for (int i = tid; i < TILE_EL; i += NTHREADS) tile[i] = 0.0f;
  __syncthreads();

  // ---- Phase 2: async copy valid interior rows, global -> LDS (ASYNCcnt) --
  // tile[r][xx] corresponds to padded-plane xp[y0 + r][xx]; data exists for
  // 1 <= (y0+r) <= 128 and 1 <= xx <= 128, sourced from x[y0+r-1][xx-1].
  for (int k = tid; k < TR_TILE * S_DIM; k += NTHREADS) {
    const int r  = k >> 7;           // 0..65 (uniform within a wave)
    const int xx = (k & 127) + 1;    // 1..128
    const int yy = y0 + r;           // padded-space row
    if (yy >= 1 && yy <= S_DIM) {
      const unsigned la = lds_addr32(&tile[r * SP + xx]);
      const float*   ga = xplane + (size_t)(yy - 1) * S_DIM + (xx - 1);
      asm volatile("global_load_async_to_lds_b32 %0, %1, off"
                   :: "v"(la), "v"(ga) : "memory");
    }
  }
#if __has_builtin(__builtin_amdgcn_s_wait_asynccnt)
  __builtin_amdgcn_s_wait_asynccnt(0);
#else
  asm volatile("s_wait_asynccnt 0" ::: "memory");
#endif
  __syncthreads();

  // ---- Phase 3: 3x3 stencil of Gaussian similarities ----------------------
  // first = 1 + E1*(4 edge taps) + E2*(4 corner taps);  (0,0) tap == exp(0).
  const float E1   = 0.6065306597126334f;   // exp(-1/2)
  const float E2   = 0.4930686913952398f;   // exp(-sqrt(2)/2)
  const float INV9 = 1.0f / 9.0f;
  const float SEC9 = 1.0729838054991615f;   // (4 + 4*sqrt(2)) / 9

  for (int g = tid; g < (TROWS * S_DIM) / 4; g += NTHREADS) {
    const int ry   = g >> 5;          // local output row 0..63
    const int xb   = (g & 31) * 4;    // column base 0..124 (float4 aligned)
    const int base = ry * SP + xb;

    // 3x6 LDS window shared by the 4 outputs of this iteration.
    float wnd[3][6];
#pragma unroll
    for (int r = 0; r < 3; ++r)
#pragma unroll
      for (int q = 0; q < 6; ++q)
        wnd[r][q] = tile[base + r * SP + q];

    const float4 v1 = *reinterpret_cast<const float4*>(w1o + 4 * g);
    const float4 v2 = *reinterpret_cast<const float4*>(w2o + 4 * g);
    const float v1a[4] = {v1.x, v1.y, v1.z, v1.w};
    const float v2a[4] = {v2.x, v2.y, v2.z, v2.w};
    float res[4];

#pragma unroll
    for (int u = 0; u < 4; ++u) {
      const float a = wnd[0][u];      // window anchor (top-left) is "center"
      float t, sE1 = 0.0f, sE2 = 0.0f;
      t = a - wnd[0][u + 1]; sE1 += __expf(-0.5f * t * t);
      t = a - wnd[0][u + 2]; sE1 += __expf(-0.5f * t * t);
      t = a - wnd[1][u];     sE1 += __expf(-0.5f * t * t);
      t = a - wnd[2][u];     sE1 += __expf(-0.5f * t * t);
      t = a - wnd[1][u + 1]; sE2 += __expf(-0.5f * t * t);
      t = a - wnd[1][u + 2]; sE2 += __expf(-0.5f * t * t);
      t = a - wnd[2][u + 1]; sE2 += __expf(-0.5f * t * t);
      t = a - wnd[2][u + 2]; sE2 += __expf(-0.5f * t * t);
      const float first = 1.0f + E1 * sE1 + E2 * sE2;
      res[u] = v1a[u] * (first * INV9) + v2a[u] * SEC9;
    }

    float4 o4 = {res[0], res[1], res[2], res[3]};
    *reinterpret_cast<float4*>(op + 4 * g) = o4;
  }
}

extern "C" void kernel_launch(void* const* d_in, const int* in_sizes, int n_in,
                              void* d_out, int out_size, void* d_ws, size_t ws_size,
                              hipStream_t stream) {
  const float* x  = (const float*)d_in[0];
  const float* w1 = (const float*)d_in[1];
  const float* w2 = (const float*)d_in[2];
  float* out = (float*)d_out;

  const int nplanes = in_sizes[0] / PLANE;   // B*C = 1024
  const int nch     = in_sizes[1] / PLANE;   // C   = 64
  if (nplanes <= 0 || nch <= 0) return;

  dim3 grid(nplanes * 2);                    // two row-halves per plane
  dim3 block(NTHREADS);
  pairwise_potential_kernel<<<grid, block, 0, stream>>>(x, w1, w2, out, nch);
}